// RangeAwareL1Loss_60009283060266
// MI455X (gfx1250) — compile-verified
//
#include <hip/hip_runtime.h>
#include <math.h>

// RangeAwareL1Loss for MI455X (gfx1250, wave32).
// Single fused streaming pass: per-bin |pred-target| sums + counts + valid count,
// conflict-free per-wave LDS histograms, WMMA f32 16x16x4 ones-matrix reduction
// epilogue, global f32 atomics into scratch, tiny finalize kernel.

#define NUM_RANGES 31
#define NB 32          // padded bin rows (row 31 stays zero)
#define WAVES 8
#define BLOCK 256
#define MAX_GRID 1024

typedef __attribute__((ext_vector_type(2))) float v2f;
typedef __attribute__((ext_vector_type(8))) float v8f;

// Scratch layout (floats): [0..31] bin sums, [32..63] bin counts, [64] total valid.
#define WS_FLOATS 65

__global__ void raw_zero_ws(float* __restrict__ ws) {
    int i = threadIdx.x;
    if (i < WS_FLOATS) ws[i] = 0.0f;
}

__global__ __launch_bounds__(BLOCK) void raw_hist_kernel(
        const float* __restrict__ pred,
        const float* __restrict__ target,
        float* __restrict__ ws,
        long long n) {
    // Per-wave private histograms: [wave][sum(0)/cnt(1)][bin 0..31][lane 0..31]
    __shared__ float h[WAVES * 2 * NB * 32];   // 64 KB
    __shared__ float vc;

    const int tid  = threadIdx.x;
    const int lane = tid & 31;
    const int wave = tid >> 5;
    float* myh = h + wave * (2 * NB * 32);

    // Zero LDS (harness does not re-poison between replays).
    for (int i = tid; i < WAVES * 2 * NB * 32; i += BLOCK) h[i] = 0.0f;
    if (tid == 0) vc = 0.0f;
    __syncthreads();

    float vcnt = 0.0f;

    // --- streaming pass: float4 (global_load_b128) grid-stride ---
    const float4* p4 = (const float4*)pred;
    const float4* t4 = (const float4*)target;
    const long long n4 = n >> 2;
    const long long gstride = (long long)gridDim.x * (long long)blockDim.x;

    auto process = [&](float p, float t) {
        const bool valid = (t != -1.0f);
        if (valid) vcnt += 1.0f;
        const float nat = expm1f(t);          // monotonic; bins in natural space
        if (valid && nat >= 0.0f) {
            int bin = (int)floorf(nat);
            if (bin > NUM_RANGES - 1) bin = NUM_RANGES - 1;
            if (bin < 0) bin = 0;
            const float base = fabsf(p - t);
            // lane-private slots -> zero intra-wave bank conflicts; ds_add_f32 no-return
            atomicAdd(&myh[bin * 32 + lane], base);
            atomicAdd(&myh[NB * 32 + bin * 32 + lane], 1.0f);
        }
    };

    for (long long i = (long long)blockIdx.x * blockDim.x + tid; i < n4; i += gstride) {
        const float4 p = p4[i];
        const float4 t = t4[i];
        process(p.x, t.x);
        process(p.y, t.y);
        process(p.z, t.z);
        process(p.w, t.w);
    }
    // Tail (n not multiple of 4): single thread mops up.
    if (blockIdx.x == 0 && tid == 0) {
        for (long long i = (n4 << 2); i < n; ++i) process(pred[i], target[i]);
    }

    atomicAdd(&vc, vcnt);
    __syncthreads();

    // --- combine the 8 wave copies into copy 0 ---
    for (int i = tid; i < 2 * NB * 32; i += BLOCK) {
        float s = h[i];
        #pragma unroll
        for (int w = 1; w < WAVES; ++w) s += h[w * (2 * NB * 32) + i];
        h[i] = s;
    }
    __syncthreads();

    if (tid == 0) atomicAdd(&ws[64], vc);

    // --- WMMA epilogue: reduce 32x32 [bin][lane] tiles across lanes with a
    // ones B-matrix. D = A(16x4) x Ones(4x16) + C accumulates row sums.
    // A layout (f32 16x4): lanes 0-15 hold (M=lane, K=k0,k0+1); lanes 16-31
    // hold (M=lane-16, K=k0+2,k0+3). 8 chained WMMAs cover K=32 columns.
    if (tid < 32) {   // wave 0 only: EXEC all ones inside this branch
        v2f ones; ones.x = 1.0f; ones.y = 1.0f;
        const int mrow = lane & 15;
        const int koff = (lane >> 4) * 2;   // 0 for lanes 0-15, 2 for lanes 16-31
        #pragma unroll
        for (int g = 0; g < 4; ++g) {
            const int arr     = g >> 1;        // 0 = sums, 1 = counts
            const int rowBase = (g & 1) * 16;  // bin rows 0-15 or 16-31
            const float* src = h + arr * (NB * 32) + (rowBase + mrow) * 32;
            v8f c = {0.f, 0.f, 0.f, 0.f, 0.f, 0.f, 0.f, 0.f};
            #pragma unroll
            for (int kb = 0; kb < 8; ++kb) {
                const int k = kb * 4 + koff;
                v2f a;
                a.x = src[k];
                a.y = src[k + 1];
                c = __builtin_amdgcn_wmma_f32_16x16x4_f32(
                        /*neg_a=*/false, a, /*neg_b=*/false, ones,
                        /*c_mod=*/(short)0, c, /*reuse_a=*/false, /*reuse_b=*/false);
            }
            // D 16x16 f32 layout: VGPR r, lane 0 -> (M=r, N=0); lane 16 -> (M=r+8, N=0)
            float* dst = ws + arr * 32 + rowBase;
            if (lane == 0) {
                #pragma unroll
                for (int m = 0; m < 8; ++m) atomicAdd(&dst[m], c[m]);
            } else if (lane == 16) {
                #pragma unroll
                for (int m = 0; m < 8; ++m) atomicAdd(&dst[8 + m], c[m]);
            }
        }
    }
}

__global__ void raw_finalize(const float* __restrict__ ws, float* __restrict__ out) {
    if (threadIdx.x == 0 && blockIdx.x == 0) {
        const float denom = fmaxf(ws[64], 1.0f);
        float loss = 0.0f;
        for (int b = 0; b < NUM_RANGES; ++b) {
            const float freq = ws[32 + b] / denom;
            const float w = 1.0f / (sqrtf(freq) + 1e-6f);  // ALPHA = 0.5
            loss += ws[b] * w;
        }
        out[0] = loss / denom;
    }
}

extern "C" void kernel_launch(void* const* d_in, const int* in_sizes, int n_in,
                              void* d_out, int out_size, void* d_ws, size_t ws_size,
                              hipStream_t stream) {
    const float* pred   = (const float*)d_in[0];
    const float* target = (const float*)d_in[1];
    float* out = (float*)d_out;
    float* ws  = (float*)d_ws;
    const long long n = (long long)in_sizes[0];

    raw_zero_ws<<<1, 128, 0, stream>>>(ws);

    const long long n4 = n >> 2;
    long long grid = (n4 + BLOCK - 1) / BLOCK;
    if (grid > MAX_GRID) grid = MAX_GRID;
    if (grid < 1) grid = 1;
    raw_hist_kernel<<<(int)grid, BLOCK, 0, stream>>>(pred, target, ws, n);

    raw_finalize<<<1, 32, 0, stream>>>(ws, out);
}